// VQ_Payam_15882789060699
// MI455X (gfx1250) — compile-verified
//
#include <hip/hip_runtime.h>
#include <math.h>

typedef float v2f __attribute__((ext_vector_type(2)));
typedef float v8f __attribute__((ext_vector_type(8)));

#define DDIM 512
#define KCODES 1024

// ---------------------------------------------------------------------------
// K0: wn[k] = -0.5 * ||W[k]||^2 ; also zero the histogram (deterministic init
// every launch). One block per code, 256 threads, float2 per thread.
// ---------------------------------------------------------------------------
__global__ __launch_bounds__(256) void vq_wnorm_kernel(
    const float* __restrict__ W, float* __restrict__ wn, int* __restrict__ hist) {
  const int k = blockIdx.x;
  const int tid = threadIdx.x;
  const float2 v = *(const float2*)(W + (size_t)k * DDIM + tid * 2);
  float s = v.x * v.x + v.y * v.y;
#pragma unroll
  for (int m = 16; m >= 1; m >>= 1) s += __shfl_down(s, m);
  __shared__ float red[8];
  const int lane = tid & 31, wid = tid >> 5;
  if (lane == 0) red[wid] = s;
  __syncthreads();
  if (tid == 0) {
    float t = 0.f;
#pragma unroll
    for (int i = 0; i < 8; ++i) t += red[i];
    wn[k] = -0.5f * t;
    hist[k] = 0;
  }
}

// ---------------------------------------------------------------------------
// K1: fused GEMM + argmax via V_WMMA_F32_16X16X4_F32.
//   score[m,n] = X[m]·W[n] - 0.5||W[n]||^2   (C preloaded with -0.5||W||^2)
//   argmin distance == argmax score.
// Block = 256 threads = 8 waves; wave w owns rows [rowBase, rowBase+16).
// Loop over 64 chunks of 16 codes; W chunk staged in LDS in B-fragment order:
//   ldsB[kc*64 + 2*lane + j] = W[colbase + (lane&15)][4*kc + 2*(lane>>4) + j]
// so the inner read is a perfectly bank-distributed ds_load_b64.
// A fragment (16x4 f32 layout): lane holds M=lane&15, K pair {4kc+2h, +1}.
// ---------------------------------------------------------------------------
__global__ __launch_bounds__(256) void vq_argmin_wmma_kernel(
    const float* __restrict__ X, const float* __restrict__ W,
    const float* __restrict__ wn, int* __restrict__ idxOut,
    int* __restrict__ hist) {
  __shared__ float ldsB[16 * DDIM];  // 32 KB, fragment-order
  const int tid = threadIdx.x;
  const int lane = tid & 31;
  const int wave = tid >> 5;
  const int nlo = lane & 15;
  const int half = lane >> 4;
  const int rowBase = blockIdx.x * 128 + wave * 16;

  const float* arow = X + (size_t)(rowBase + nlo) * DDIM + 2 * half;

  float best[8];
  int bidx[8];
#pragma unroll
  for (int i = 0; i < 8; ++i) { best[i] = -3.402823466e38f; bidx[i] = 0; }

  const int fcol = tid & 15;   // code within chunk this thread stages
  const int fkq = tid >> 4;    // 0..15 -> k group

  for (int cc = 0; cc < KCODES / 16; ++cc) {
    const int colbase = cc * 16;
    __syncthreads();
    {
      const float* wrow = W + (size_t)(colbase + fcol) * DDIM;
#pragma unroll
      for (int it = 0; it < 8; ++it) {
        const int k0 = fkq * 4 + it * 64;               // multiple of 4
        const float4 w4 = *(const float4*)(wrow + k0);  // 16B aligned
        const int e0 = (k0 >> 2) * 64 + 2 * fcol;
        *(float2*)(&ldsB[e0]) = make_float2(w4.x, w4.y);        // h=0: j=0,1
        *(float2*)(&ldsB[e0 + 32]) = make_float2(w4.z, w4.w);   // h=1: j=0,1
      }
    }
    __syncthreads();

    const float wnv = wn[colbase + nlo];
    v8f c = {wnv, wnv, wnv, wnv, wnv, wnv, wnv, wnv};
#pragma unroll 8
    for (int kc = 0; kc < DDIM / 4; ++kc) {
      const v2f a = *(const v2f*)(arow + 4 * kc);
      const v2f b = *(const v2f*)(&ldsB[kc * 64 + 2 * lane]);
      c = __builtin_amdgcn_wmma_f32_16x16x4_f32(
          /*neg_a=*/false, a, /*neg_b=*/false, b,
          /*c_mod=*/(short)0, c, /*reuse_a=*/false, /*reuse_b=*/false);
    }

    const int colIdx = colbase + nlo;
#pragma unroll
    for (int i = 0; i < 8; ++i) {
      if (c[i] > best[i]) { best[i] = c[i]; bidx[i] = colIdx; }  // tie -> keep earlier
    }
  }

  // Argmax tree across the 16-lane group holding one row's 16 column classes.
  // C layout: reg i, lanes 0-15 -> row rowBase+i ; lanes 16-31 -> rowBase+8+i.
#pragma unroll
  for (int m = 8; m >= 1; m >>= 1) {
#pragma unroll
    for (int i = 0; i < 8; ++i) {
      const float ov = __shfl_xor(best[i], m);
      const int oi = __shfl_xor(bidx[i], m);
      if (ov > best[i] || (ov == best[i] && oi < bidx[i])) {
        best[i] = ov; bidx[i] = oi;
      }
    }
  }
  if (nlo == 0) {  // lanes 0 and 16
    const int mbase = rowBase + half * 8;
#pragma unroll
    for (int i = 0; i < 8; ++i) {
      idxOut[mbase + i] = bidx[i];
      atomicAdd(&hist[bidx[i]], 1);  // integer: deterministic
    }
  }
}

// ---------------------------------------------------------------------------
// K2: gather codebook rows -> quantized_st, one-hot encodings, partial SSE.
// out layout: [0]=loss, [1 .. 1+Q)=quantized_st, [1+Q]=perplexity,
//             [2+Q .. 2+Q+N*K)=encodings, Q = nrows*DDIM.
// ---------------------------------------------------------------------------
__global__ __launch_bounds__(256) void vq_gather_kernel(
    const float* __restrict__ X, const float* __restrict__ W,
    const int* __restrict__ idx, float* __restrict__ out, size_t qElems,
    float* __restrict__ partials) {
  const int tid = threadIdx.x;
  const int b = blockIdx.x;
  float acc = 0.f;
#pragma unroll
  for (int r = 0; r < 4; ++r) {
    const int row = b * 4 + r;
    const int ci = idx[row];
    const float* wrow = W + (size_t)ci * DDIM;
    const float* xrow = X + (size_t)row * DDIM;
    float* qrow = out + 1 + (size_t)row * DDIM;
#pragma unroll
    for (int e = 0; e < 2; ++e) {
      const int d = tid + e * 256;
      const float q = wrow[d];
      const float xv = xrow[d];
      qrow[d] = xv + (q - xv);  // straight-through, exactly as reference
      const float diff = q - xv;
      acc += diff * diff;
    }
    float* erow = out + 2 + qElems + (size_t)row * KCODES;
    const int base = tid * 4;
    float2 z0 = make_float2(0.f, 0.f), z1 = make_float2(0.f, 0.f);
    const int rel = ci - base;
    if (rel == 0) z0.x = 1.f;
    else if (rel == 1) z0.y = 1.f;
    else if (rel == 2) z1.x = 1.f;
    else if (rel == 3) z1.y = 1.f;
    *(float2*)(erow + base) = z0;       // 8B aligned (offset even)
    *(float2*)(erow + base + 2) = z1;
  }
  __shared__ float red[256];
  red[tid] = acc;
  __syncthreads();
#pragma unroll
  for (int off = 128; off >= 1; off >>= 1) {
    if (tid < off) red[tid] += red[tid + off];
    __syncthreads();
  }
  if (tid == 0) partials[b] = red[0];
}

// ---------------------------------------------------------------------------
// K3: deterministic final reductions: loss and perplexity.
// ---------------------------------------------------------------------------
__global__ __launch_bounds__(256) void vq_finalize_kernel(
    const float* __restrict__ partials, int nPart, const int* __restrict__ hist,
    float* __restrict__ out, size_t qElems, float invRows) {
  __shared__ float red[256];
  const int t = threadIdx.x;
  float s = 0.f;
  for (int i = t; i < nPart; i += 256) s += partials[i];  // fixed order
  red[t] = s;
  __syncthreads();
#pragma unroll
  for (int off = 128; off >= 1; off >>= 1) {
    if (t < off) red[t] += red[t + off];
    __syncthreads();
  }
  if (t == 0) out[0] = 1.25f * red[0] / (float)qElems;  // q_loss + 0.25*e_loss
  __syncthreads();

  float e = 0.f;
  for (int i = t; i < KCODES; i += 256) {
    const float p = (float)hist[i] * invRows;
    e += p * logf(p + 1e-10f);
  }
  red[t] = e;
  __syncthreads();
#pragma unroll
  for (int off = 128; off >= 1; off >>= 1) {
    if (t < off) red[t] += red[t + off];
    __syncthreads();
  }
  if (t == 0) out[1 + qElems] = expf(-red[0]);
}

// ---------------------------------------------------------------------------
extern "C" void kernel_launch(void* const* d_in, const int* in_sizes, int n_in,
                              void* d_out, int out_size, void* d_ws,
                              size_t ws_size, hipStream_t stream) {
  const float* X = (const float*)d_in[0];  // [128,512,512] f32
  const float* W = (const float*)d_in[1];  // [1024,512] f32
  float* out = (float*)d_out;

  const int nrows = in_sizes[0] / DDIM;       // 65536
  const size_t qElems = (size_t)nrows * DDIM; // 33554432

  // workspace layout
  float* ws_wn = (float*)d_ws;              // KCODES floats
  int* ws_idx = (int*)(ws_wn + KCODES);     // nrows ints
  int* ws_hist = ws_idx + nrows;            // KCODES ints
  float* ws_part = (float*)(ws_hist + KCODES);  // nrows/4 floats

  vq_wnorm_kernel<<<KCODES, 256, 0, stream>>>(W, ws_wn, ws_hist);
  vq_argmin_wmma_kernel<<<nrows / 128, 256, 0, stream>>>(X, W, ws_wn, ws_idx,
                                                         ws_hist);
  const int nPart = nrows / 4;
  vq_gather_kernel<<<nPart, 256, 0, stream>>>(X, W, ws_idx, out, qElems,
                                              ws_part);
  vq_finalize_kernel<<<1, 256, 0, stream>>>(ws_part, nPart, ws_hist, out,
                                            qElems, 1.0f / (float)nrows);
}